// FCOSBBox_30236569764336
// MI455X (gfx1250) — compile-verified
//
#include <hip/hip_runtime.h>
#include <hip/hip_bf16.h>

// ---------------- problem constants (from the reference) ----------------
#define NUM_CLASSES 80
#define BATCH       4
#define KTOT        4720          // 1000*4 + 720
#define LOGIT_THR  (-3.6635616f)  // logit(0.025): sigmoid(x)>0.025 <=> x>this
#define SCORE_THR   0.025f
#define IMG_MAX     320.0f
#define IOU_THR     0.5f
#define NPAD_SEL    1024          // per-level top-k sort width (k<=1000)
#define NPAD_SORT   8192          // global sort width (K=4720)

// ---------------- CDNA5 TDM (tensor_load_to_lds) ----------------
typedef unsigned int u32x4  __attribute__((ext_vector_type(4)));
typedef int          i32x4v __attribute__((ext_vector_type(4)));
typedef int          i32x8v __attribute__((ext_vector_type(8)));

#if defined(__AMDGCN__)
#  if __has_builtin(__builtin_amdgcn_tensor_load_to_lds)
#    define HAVE_TDM 1
#  endif
#endif
#if __has_include(<hip/amd_detail/amd_gfx1250_TDM.h>)
#  define TDM_ARITY6 1   // amdgpu-toolchain (therock) 6-arg builtin
#endif

#ifdef HAVE_TDM
// Flat 1-D tile copy: global -> LDS, nElems fp32 elements (nElems < 65536).
// D# group0: count=1, lds_addr, global_addr[56:0], type=2 ("image").
// D# group1: data_size=4B, tensor_dim0=tile_dim0=nElems, dims1/2 unused.
__device__ __forceinline__ void tdm_load_1d(const void* gsrc, void* ldst,
                                            unsigned nElems) {
  unsigned ldsAddr = (unsigned)(unsigned long long)ldst; // flat low 32b == LDS offset
  unsigned long long ga = (unsigned long long)gsrc;
  u32x4 g0;
  g0.x = 1u;                                             // count=1, user descriptor
  g0.y = ldsAddr;                                        // lds_addr
  g0.z = (unsigned)(ga & 0xFFFFFFFFull);                 // global_addr[31:0]
  g0.w = (unsigned)((ga >> 32) & 0x01FFFFFFull)          // global_addr[56:32]
       | (2u << 30);                                     // type=2
  i32x8v g1;
  g1[0] = (int)(2u << 16);                               // data_size=4B, mask=0
  g1[1] = (int)((nElems & 0xFFFFu) << 16);               // tensor_dim0[15:0]
  g1[2] = (int)(((nElems >> 16) & 0xFFFFu) | (1u << 16));// tensor_dim0[31:16], tdim1=1
  g1[3] = (int)((nElems & 0xFFFFu) << 16);               // tile_dim0
  g1[4] = 1;                                             // tile_dim1=1, tile_dim2=0
  g1[5] = (int)nElems;                                   // tensor_dim0_stride[31:0]
  g1[6] = 0;
  g1[7] = 0;
  i32x4v z4 = {0, 0, 0, 0};
#ifdef TDM_ARITY6
  i32x8v z8 = {0, 0, 0, 0, 0, 0, 0, 0};
  __builtin_amdgcn_tensor_load_to_lds(g0, g1, z4, z4, z8, 0);
#else
  __builtin_amdgcn_tensor_load_to_lds(g0, g1, z4, z4, 0);
#endif
}
#endif // HAVE_TDM

// ---------------- helpers ----------------
// Order-preserving float -> u32 (larger float => larger key), valid for all finite.
__device__ __forceinline__ unsigned fkey(float f) {
  unsigned u = __float_as_uint(f);
  return (u & 0x80000000u) ? ~u : (u | 0x80000000u);
}

// Composite selection key: (masked-score key << 32) | ~idx.
// Unique per candidate => radix-select is exact; ties in score resolve to the
// LOWEST index first, matching jax.lax.top_k.
__device__ __forceinline__ unsigned long long cand_key(
    const float* __restrict__ cls, int hw, int t) {
  int c = t % NUM_CLASSES;
  int a = t / NUM_CLASSES;
  float logit = cls[c * hw + a];                // cls[b,c,y,x], a = y*w+x
  unsigned k32 = (logit > LOGIT_THR) ? fkey(logit) : 0u; // masked -> smallest
  return ((unsigned long long)k32 << 32) |
         (unsigned long long)(0xFFFFFFFFu - (unsigned)t);
}

__device__ __forceinline__ float warp_max32(float m) {
  #pragma unroll
  for (int o = 16; o > 0; o >>= 1) m = fmaxf(m, __shfl_xor(m, o, 32));
  return m;
}

// =====================================================================
// Stage A: per (image, level) — sigmoid/threshold/top-k/decode.
// One 1024-thread block per (b,lvl). Exact top-k via 64-bit radix-select.
// =====================================================================
__global__ __launch_bounds__(1024)
void fcos_select(const float* __restrict__ cls0, const float* __restrict__ cls1,
                 const float* __restrict__ cls2, const float* __restrict__ cls3,
                 const float* __restrict__ cls4,
                 const float* __restrict__ bb0, const float* __restrict__ bb1,
                 const float* __restrict__ bb2, const float* __restrict__ bb3,
                 const float* __restrict__ bb4,
                 const float* __restrict__ ct0, const float* __restrict__ ct1,
                 const float* __restrict__ ct2, const float* __restrict__ ct3,
                 const float* __restrict__ ct4,
                 float* __restrict__ rawScore, int* __restrict__ rawLabel,
                 float* __restrict__ rawBox) {
  const int hs[5] = {40, 20, 10, 5, 3};
  const int blk = blockIdx.x;
  const int b = blk / 5, lvl = blk % 5;
  const int tid = threadIdx.x, NT = blockDim.x;
  const int h = hs[lvl], w = h, hw = h * w;
  const int N = hw * NUM_CLASSES;
  const int klvl = (N < 1000) ? N : 1000;
  const int lvlOff = lvl * 1000;
  const float stride = (float)(8 << lvl);

  const float* const clsA[5] = {cls0, cls1, cls2, cls3, cls4};
  const float* const bbA[5]  = {bb0, bb1, bb2, bb3, bb4};
  const float* const ctA[5]  = {ct0, ct1, ct2, ct3, ct4};
  const float* cls = clsA[lvl] + (size_t)b * NUM_CLASSES * hw;
  const float* bb  = bbA[lvl] + (size_t)b * 4 * hw;
  const float* ct  = ctA[lvl] + (size_t)b * hw;

  __shared__ float sBB[4 * 1600];   // ltrb planes, level-0 worst case
  __shared__ float sCtr[1600];
  __shared__ unsigned sHist[256];
  __shared__ unsigned long long sKeyBuf[NPAD_SEL];
  __shared__ unsigned long long sPrefix;
  __shared__ unsigned sRemain, sCnt;

  __builtin_prefetch(cls + (size_t)tid * 32, 0, 0);  // global_prefetch_b8

  // --- stage bbox + centerness tiles into LDS via the Tensor Data Mover ---
#ifdef HAVE_TDM
  if (tid < 32) {                 // one wave issues; EXEC ignored by TDM
    tdm_load_1d(bb, sBB, (unsigned)(4 * hw));
    tdm_load_1d(ct, sCtr, (unsigned)hw);
    __builtin_amdgcn_s_wait_tensorcnt(0);
  }
#else
  for (int t = tid; t < 4 * hw; t += NT) sBB[t] = bb[t];
  for (int t = tid; t < hw; t += NT)     sCtr[t] = ct[t];
#endif
  if (tid == 0) { sPrefix = 0ull; sRemain = (unsigned)klvl; }
  __syncthreads();

  // --- 64-bit radix-select: find k-th largest composite key ---
  for (int d = 7; d >= 0; --d) {
    for (int t = tid; t < 256; t += NT) sHist[t] = 0u;
    __syncthreads();
    unsigned long long pfx = sPrefix;
    unsigned long long hmask = (d == 7) ? 0ull : (~0ull << (8 * (d + 1)));
    for (int t = tid; t < N; t += NT) {
      unsigned long long ck = cand_key(cls, hw, t);
      if ((ck & hmask) == (pfx & hmask)) {
        unsigned dig = (unsigned)((ck >> (8 * d)) & 0xFFull);
        atomicAdd(&sHist[dig], 1u);
      }
    }
    __syncthreads();
    if (tid == 0) {
      unsigned rem = sRemain, cum = 0;
      int v = 255;
      for (; v > 0; --v) { cum += sHist[v]; if (cum >= rem) break; }
      if (cum < rem) cum += sHist[0];              // v==0 fallthrough
      sPrefix = pfx | ((unsigned long long)(unsigned)v << (8 * d));
      sRemain = rem - (cum - sHist[v]);
    }
    __syncthreads();
  }

  // --- compact the k survivors (keys unique => exactly klvl of them) ---
  unsigned long long T = sPrefix;
  if (tid == 0) sCnt = 0u;
  __syncthreads();
  for (int t = tid; t < N; t += NT) {
    unsigned long long ck = cand_key(cls, hw, t);
    if (ck >= T) {
      unsigned pos = atomicAdd(&sCnt, 1u);
      if (pos < NPAD_SEL) sKeyBuf[pos] = ck;
    }
  }
  __syncthreads();
  unsigned cnt = sCnt;
  for (int t = tid; t < NPAD_SEL; t += NT)
    if (t >= (int)cnt) sKeyBuf[t] = 0ull;
  __syncthreads();

  // --- bitonic sort 1024 keys, descending ---
  for (int kk = 2; kk <= NPAD_SEL; kk <<= 1) {
    for (int j = kk >> 1; j > 0; j >>= 1) {
      int t = tid, ixj = t ^ j;
      if (ixj > t) {
        unsigned long long a = sKeyBuf[t], c = sKeyBuf[ixj];
        bool desc = ((t & kk) == 0);
        if (desc ? (a < c) : (a > c)) { sKeyBuf[t] = c; sKeyBuf[ixj] = a; }
      }
      __syncthreads();
    }
  }

  // --- decode the top-k entries (boxes computed even for masked entries) ---
  if (tid < klvl) {
    unsigned long long ck = sKeyBuf[tid];
    unsigned idx = 0xFFFFFFFFu - (unsigned)(ck & 0xFFFFFFFFull);
    int o = b * KTOT + lvlOff + tid;
    if (idx < (unsigned)N) {
      int c = (int)(idx % NUM_CLASSES);
      int a = (int)(idx / NUM_CLASSES);
      int y = a / w, x = a % w;
      float logit = cls[c * hw + a];
      float s  = 1.0f / (1.0f + expf(-logit));
      float sf = 1.0f / (1.0f + expf(-sCtr[a]));
      bool valid = logit > LOGIT_THR;                  // == (s > SCORE_THR)
      float px = ((float)x + 0.5f) * stride;
      float py = ((float)y + 0.5f) * stride;
      float x1 = fminf(fmaxf(px - sBB[0 * hw + a], 0.0f), IMG_MAX);
      float y1 = fminf(fmaxf(py - sBB[1 * hw + a], 0.0f), IMG_MAX);
      float x2 = fminf(fmaxf(px + sBB[2 * hw + a], 0.0f), IMG_MAX);
      float y2 = fminf(fmaxf(py + sBB[3 * hw + a], 0.0f), IMG_MAX);
      rawScore[o] = valid ? (s * sf) : -1.0f;
      rawLabel[o] = c;
      rawBox[(size_t)o * 4 + 0] = x1;
      rawBox[(size_t)o * 4 + 1] = y1;
      rawBox[(size_t)o * 4 + 2] = x2;
      rawBox[(size_t)o * 4 + 3] = y2;
    } else {                                           // defensive padding
      rawScore[o] = -1.0f; rawLabel[o] = 0;
      rawBox[(size_t)o * 4 + 0] = 0.f; rawBox[(size_t)o * 4 + 1] = 0.f;
      rawBox[(size_t)o * 4 + 2] = 0.f; rawBox[(size_t)o * 4 + 3] = 0.f;
    }
  }
}

// =====================================================================
// Stage B: per image — stable sort K=4720 by score desc; gather to d_out.
// Dynamic LDS: 8192 x u64 composite keys (score-key<<32 | ~idx).
// =====================================================================
__global__ __launch_bounds__(1024)
void fcos_sort(const float* __restrict__ rawScore, const int* __restrict__ rawLabel,
               const float* __restrict__ rawBox,
               float* __restrict__ outDets, float* __restrict__ outLabels) {
  extern __shared__ unsigned long long sk[];
  const int b = blockIdx.x, tid = threadIdx.x, NT = blockDim.x;
  for (int t = tid; t < NPAD_SORT; t += NT) {
    unsigned long long key = 0ull;
    if (t < KTOT) {
      float s = rawScore[b * KTOT + t];
      key = ((unsigned long long)fkey(s) << 32) |
            (unsigned long long)(0xFFFFFFFFu - (unsigned)t);
    }
    sk[t] = key;
  }
  __syncthreads();
  for (int kk = 2; kk <= NPAD_SORT; kk <<= 1) {
    for (int j = kk >> 1; j > 0; j >>= 1) {
      for (int t = tid; t < NPAD_SORT; t += NT) {
        int ixj = t ^ j;
        if (ixj > t) {
          unsigned long long a = sk[t], c = sk[ixj];
          bool desc = ((t & kk) == 0);
          if (desc ? (a < c) : (a > c)) { sk[t] = c; sk[ixj] = a; }
        }
      }
      __syncthreads();
    }
  }
  for (int r = tid; r < KTOT; r += NT) {
    unsigned j = 0xFFFFFFFFu - (unsigned)(sk[r] & 0xFFFFFFFFull);
    size_t src = (size_t)b * KTOT + j, dst = (size_t)b * KTOT + r;
    outDets[dst * 5 + 0] = rawBox[src * 4 + 0];
    outDets[dst * 5 + 1] = rawBox[src * 4 + 1];
    outDets[dst * 5 + 2] = rawBox[src * 4 + 2];
    outDets[dst * 5 + 3] = rawBox[src * 4 + 3];
    outDets[dst * 5 + 4] = rawScore[src];
    outLabels[dst] = (float)rawLabel[src];
  }
}

// =====================================================================
// Stage C: per image — class-offset boxes, greedy NMS, keep -> d_out.
// No K x K matrix: sequential over i (score order), parallel over j.
// =====================================================================
__global__ __launch_bounds__(1024)
void fcos_nms(const float* __restrict__ outDets, const float* __restrict__ outLabels,
              float* __restrict__ offBox, float* __restrict__ area,
              float* __restrict__ outKeep) {
  __shared__ int sKeep[KTOT];
  __shared__ float sRed[32];
  const int b = blockIdx.x, tid = threadIdx.x, NT = blockDim.x;
  const float* dets = outDets + (size_t)b * KTOT * 5;
  float* ob = offBox + (size_t)b * KTOT * 4;
  float* ar = area + (size_t)b * KTOT;

  // 1) max over all box coordinates (wave32 shuffle + LDS tree)
  float m = -3.4e38f;
  for (int t = tid; t < KTOT * 4; t += NT)
    m = fmaxf(m, dets[(size_t)(t >> 2) * 5 + (t & 3)]);
  m = warp_max32(m);
  if ((tid & 31) == 0) sRed[tid >> 5] = m;
  __syncthreads();
  if (tid < 32) {
    float v = (tid < (NT >> 5)) ? sRed[tid] : -3.4e38f;
    v = warp_max32(v);
    if (tid == 0) sRed[0] = v;
  }
  __syncthreads();
  const float off = sRed[0] + 1.0f;

  // 2) per-class offset boxes, areas, keep init (valid <=> score > 0)
  for (int i = tid; i < KTOT; i += NT) {
    float d = outLabels[(size_t)b * KTOT + i] * off;
    float x1 = dets[(size_t)i * 5 + 0] + d, y1 = dets[(size_t)i * 5 + 1] + d;
    float x2 = dets[(size_t)i * 5 + 2] + d, y2 = dets[(size_t)i * 5 + 3] + d;
    ob[(size_t)i * 4 + 0] = x1; ob[(size_t)i * 4 + 1] = y1;
    ob[(size_t)i * 4 + 2] = x2; ob[(size_t)i * 4 + 3] = y2;
    ar[i] = (x2 - x1) * (y2 - y1);
    sKeep[i] = (dets[(size_t)i * 5 + 4] > 0.0f) ? 1 : 0;
  }
  __syncthreads();

  // 3) greedy suppression in score order
  for (int i = 0; i < KTOT; ++i) {
    if (sKeep[i]) {
      float x1i = ob[(size_t)i * 4 + 0], y1i = ob[(size_t)i * 4 + 1];
      float x2i = ob[(size_t)i * 4 + 2], y2i = ob[(size_t)i * 4 + 3];
      float ai = ar[i];
      for (int j = i + 1 + tid; j < KTOT; j += NT) {
        float xx1 = fmaxf(x1i, ob[(size_t)j * 4 + 0]);
        float yy1 = fmaxf(y1i, ob[(size_t)j * 4 + 1]);
        float xx2 = fminf(x2i, ob[(size_t)j * 4 + 2]);
        float yy2 = fminf(y2i, ob[(size_t)j * 4 + 3]);
        float iw = fmaxf(xx2 - xx1, 0.0f), ih = fmaxf(yy2 - yy1, 0.0f);
        float inter = iw * ih;
        float uni = ai + ar[j] - inter;
        float iou = (uni > 0.0f) ? (inter / uni) : 0.0f;
        if (iou > IOU_THR) sKeep[j] = 0;
      }
    }
    __syncthreads();
  }
  for (int i = tid; i < KTOT; i += NT)
    outKeep[(size_t)b * KTOT + i] = (float)sKeep[i];
}

// =====================================================================
extern "C" void kernel_launch(void* const* d_in, const int* in_sizes, int n_in,
                              void* d_out, int out_size, void* d_ws, size_t ws_size,
                              hipStream_t stream) {
  (void)in_sizes; (void)n_in; (void)out_size; (void)ws_size;
  // setup_inputs() dict order: cls0, bbox0, ctr0, cls1, bbox1, ctr1, ...
  const float* cls[5]; const float* bbx[5]; const float* ctr[5];
  for (int l = 0; l < 5; ++l) {
    cls[l] = (const float*)d_in[3 * l + 0];
    bbx[l] = (const float*)d_in[3 * l + 1];
    ctr[l] = (const float*)d_in[3 * l + 2];
  }
  float* ws = (float*)d_ws;
  const size_t BK = (size_t)BATCH * KTOT;
  float* rawScore = ws;                       // BK
  int*   rawLabel = (int*)(ws + BK);          // BK
  float* rawBox   = ws + 2 * BK;              // BK*4
  float* offBox   = ws + 6 * BK;              // BK*4
  float* area     = ws + 10 * BK;             // BK    (total 11*BK floats)

  float* out = (float*)d_out;
  float* outDets   = out;                     // (B,K,5)
  float* outLabels = out + BK * 5;            // (B,K)
  float* outKeep   = out + BK * 6;            // (B,K)

  fcos_select<<<BATCH * 5, 1024, 0, stream>>>(
      cls[0], cls[1], cls[2], cls[3], cls[4],
      bbx[0], bbx[1], bbx[2], bbx[3], bbx[4],
      ctr[0], ctr[1], ctr[2], ctr[3], ctr[4],
      rawScore, rawLabel, rawBox);
  fcos_sort<<<BATCH, 1024, NPAD_SORT * sizeof(unsigned long long), stream>>>(
      rawScore, rawLabel, rawBox, outDets, outLabels);
  fcos_nms<<<BATCH, 1024, 0, stream>>>(outDets, outLabels, offBox, area, outKeep);
}